// AttentionSinkRope_75977971466861
// MI455X (gfx1250) — compile-verified
//
#include <hip/hip_runtime.h>
#include <hip/hip_bf16.h>
#include <stdint.h>

// ---- problem constants (from reference) ----
#define NLAYERS   16
#define NHEADS    8
#define HEAD_DIM  128
#define CACHE_LEN 4064                 // MAX_CTX - AR_LEN
#define SINK      4
#define ORIG_POS  132                  // SINK + EVICT
#define KEEP      3932                 // 4096 - 4 - 128 - 32
#define SLICES    (NLAYERS * NHEADS)   // 128 (B == 1)
#define SLICE_ELEMS ((size_t)HEAD_DIM * CACHE_LEN)   // 520192 floats per slice
#define K_TOTAL   ((size_t)SLICES * SLICE_ELEMS)     // 66,584,576 floats

typedef float v4f __attribute__((ext_vector_type(4)));

// ---------------------------------------------------------------------------
// K: re-rotate kept columns (complex rotation across head-dim pairs),
// copy sink columns, zero evicted tail columns.
// Streaming payload uses non-temporal b128 accesses (read/written exactly
// once; keeps the 2 MB cos/sin table resident in L2 instead).
// Grid: (4 position-chunks of 1024, 64 dim-pairs, 128 slices), 256 thr/blk.
// ---------------------------------------------------------------------------
__global__ __launch_bounds__(256) void k_rerope_kernel(
    const float* __restrict__ kin, float* __restrict__ kout,
    const float* __restrict__ rcos, const float* __restrict__ rsin)
{
    const int tid = threadIdx.x;
    const int p   = blockIdx.x * 1024 + tid * 4;   // position within KEEP window
    const int c   = blockIdx.y;                    // head-dim pair index (0..63)
    const int s   = blockIdx.z;                    // layer*head slice (0..127)

    const size_t base = (size_t)s * SLICE_ELEMS + (size_t)(2 * c) * CACHE_LEN;
    const float* in0 = kin  + base;                // row d = 2c
    const float* in1 = in0  + CACHE_LEN;           // row d = 2c+1
    float*       o0  = kout + base;
    float*       o1  = o0   + CACHE_LEN;

    if (p < KEEP) {
        v4f a = __builtin_nontemporal_load((const v4f*)(in0 + ORIG_POS + p));
        v4f b = __builtin_nontemporal_load((const v4f*)(in1 + ORIG_POS + p));
        const float* cb = rcos + (size_t)p * 64 + c;   // (KEEP, 64) row-major
        const float* sb = rsin + (size_t)p * 64 + c;
        float c0 = cb[0], c1 = cb[64], c2 = cb[128], c3 = cb[192];
        float s0 = sb[0], s1 = sb[64], s2 = sb[128], s3 = sb[192];
        v4f r, i;
        r.x = a.x * c0 - b.x * s0;  i.x = a.x * s0 + b.x * c0;
        r.y = a.y * c1 - b.y * s1;  i.y = a.y * s1 + b.y * c1;
        r.z = a.z * c2 - b.z * s2;  i.z = a.z * s2 + b.z * c2;
        r.w = a.w * c3 - b.w * s3;  i.w = a.w * s3 + b.w * c3;
        __builtin_nontemporal_store(r, (v4f*)(o0 + SINK + p));
        __builtin_nontemporal_store(i, (v4f*)(o1 + SINK + p));
    } else if (p < SINK + CACHE_LEN - 4) {         // out cols 3936..4063 -> zero
        v4f z = {0.f, 0.f, 0.f, 0.f};
        __builtin_nontemporal_store(z, (v4f*)(o0 + SINK + p));
        __builtin_nontemporal_store(z, (v4f*)(o1 + SINK + p));
    } else if (tid == 255) {                       // one idle lane: sink columns 0..3
        v4f a = __builtin_nontemporal_load((const v4f*)in0);
        v4f b = __builtin_nontemporal_load((const v4f*)in1);
        __builtin_nontemporal_store(a, (v4f*)o0);
        __builtin_nontemporal_store(b, (v4f*)o1);
    }
}

// ---------------------------------------------------------------------------
// V: pure row-shift copy via CDNA5 async global<->LDS DMA (ASYNCcnt path),
// plus zero-fill of the evicted tail rows. Non-temporal hints on the DMA.
// Per slice: copy float4 q in [0,125952): src = q<128 ? q : q+4096 ; dst = q.
// 125952 = 41 blocks * 3 rounds * 4 buffers * 256 lanes (exact).
// Grid: (41, 128), 256 thr/blk, 16 KB LDS staging.
// ---------------------------------------------------------------------------
#define V_COPY4  125952
#define V_SINK4  128
#define V_SHIFT4 4096     // (ORIG_POS - SINK) * HEAD_DIM / 4
#define V_ROUNDS 3

__global__ __launch_bounds__(256) void v_copy_async_kernel(
    const float* __restrict__ vin, float* __restrict__ vout)
{
    __shared__ alignas(16) float smem[4 * 256 * 4];   // 4 pipeline buffers, 16 KB
    const int tid = threadIdx.x;
    const int s   = blockIdx.y;

    const v4f* in4  = (const v4f*)(vin  + (size_t)s * SLICE_ELEMS);
    v4f*       out4 = (v4f*)      (vout + (size_t)s * SLICE_ELEMS);

    // LDS byte offsets for this lane's four 16B slots (low 32 bits of the
    // generic pointer to a __shared__ object == LDS offset on amdgcn).
    const unsigned lbase = (unsigned)(uint64_t)(void*)&smem[0];
    unsigned loff[4];
#pragma unroll
    for (int b = 0; b < 4; ++b) loff[b] = lbase + (unsigned)((b * 256 + tid) * 16);

    int q0 = blockIdx.x * (V_ROUNDS * 1024) + tid;
    for (int r = 0; r < V_ROUNDS; ++r, q0 += 1024) {
#pragma unroll
        for (int b = 0; b < 4; ++b) {
            const int q = q0 + b * 256;
            const v4f* src = in4 + (q < V_SINK4 ? q : q + V_SHIFT4);
            asm volatile("global_load_async_to_lds_b128 %0, %1, off th:TH_LOAD_NT"
                         :: "v"(loff[b]), "v"(src) : "memory");
        }
        asm volatile("s_wait_asynccnt 0x0" ::: "memory");   // loads landed in LDS
#pragma unroll
        for (int b = 0; b < 4; ++b) {
            v4f* dst = out4 + (q0 + b * 256);
            asm volatile("global_store_async_from_lds_b128 %0, %1, off th:TH_STORE_NT"
                         :: "v"(dst), "v"(loff[b]) : "memory");
        }
        asm volatile("s_wait_asynccnt 0x0" ::: "memory");   // LDS safe to reuse
    }

    // Zero evicted tail rows: out4[125952 .. 130048), 4096 float4 per slice.
    if (blockIdx.x < 16) {
        v4f z = {0.f, 0.f, 0.f, 0.f};
        __builtin_nontemporal_store(z, &out4[V_COPY4 + blockIdx.x * 256 + tid]);
    }
}

// ---------------------------------------------------------------------------
// Host launcher. Inputs: k_caches, v_caches, rerotation_cos, rerotation_sin.
// d_out = new_k (flat) then new_v (flat).
// ---------------------------------------------------------------------------
extern "C" void kernel_launch(void* const* d_in, const int* in_sizes, int n_in,
                              void* d_out, int out_size, void* d_ws, size_t ws_size,
                              hipStream_t stream)
{
    (void)in_sizes; (void)n_in; (void)out_size; (void)d_ws; (void)ws_size;

    const float* k_in = (const float*)d_in[0];
    const float* v_in = (const float*)d_in[1];
    const float* rcos = (const float*)d_in[2];
    const float* rsin = (const float*)d_in[3];

    float* out_k = (float*)d_out;
    float* out_v = out_k + K_TOTAL;

    k_rerope_kernel<<<dim3(4, 64, 128), 256, 0, stream>>>(k_in, out_k, rcos, rsin);
    v_copy_async_kernel<<<dim3(41, 128), 256, 0, stream>>>(v_in, out_v);
}